// Contrastive_31628139168312
// MI455X (gfx1250) — compile-verified
//
#include <hip/hip_runtime.h>
#include <math.h>

#define N_NODES 8192
#define DIM     16
#define KNN     10
#define MARGIN  0.1f
#define EPSF    1e-12f
#define N_SIG   40000
#define N_RAND  100000

typedef __attribute__((ext_vector_type(2))) float v2f;
typedef __attribute__((ext_vector_type(8))) float v8f;

// ---------------------------------------------------------------- utilities

__global__ void init_acc_kernel(float* __restrict__ acc) {
    if (threadIdx.x < 4) acc[threadIdx.x] = 0.0f;
}

// per-node squared norm: sq[i] = dot(emb[i], emb[i])
__global__ void sq_kernel(const float* __restrict__ emb, float* __restrict__ sq) {
    int i = blockIdx.x * blockDim.x + threadIdx.x;
    if (i < N_NODES) {
        const float4* r = (const float4*)(emb + i * DIM);
        float s = 0.f;
#pragma unroll
        for (int c = 0; c < 4; ++c) {
            float4 v = r[c];
            s += v.x * v.x + v.y * v.y + v.z * v.z + v.w * v.w;
        }
        sq[i] = s;
    }
}

// ---------------------------------------------------------------- edge losses

__global__ void edge_loss_kernel(const float* __restrict__ emb,
                                 const int* __restrict__ pid,
                                 const int* __restrict__ src,
                                 const int* __restrict__ dst,
                                 int nedges, int force_pos,
                                 float* __restrict__ acc) {
    int e = blockIdx.x * blockDim.x + threadIdx.x;
    float per = 0.f;
    if (e < nedges) {
        int s = src[e], d = dst[e];
        const float4* rs = (const float4*)(emb + s * DIM);
        const float4* rd = (const float4*)(emb + d * DIM);
        float d2 = 0.f;
#pragma unroll
        for (int c = 0; c < 4; ++c) {
            float4 a = rs[c], b = rd[c];
            float dx = a.x - b.x, dy = a.y - b.y, dz = a.z - b.z, dw = a.w - b.w;
            d2 += dx * dx + dy * dy + dz * dz + dw * dw;
        }
        float dist = sqrtf(d2 + EPSF);
        bool y = force_pos ? true : ((pid[s] == pid[d]) && (pid[s] != 0));
        per = y ? dist : fmaxf(0.f, MARGIN - dist);
    }
    // wave32 reduction, one atomic per wave
#pragma unroll
    for (int off = 16; off > 0; off >>= 1)
        per += __shfl_down(per, off, 32);
    if ((threadIdx.x & 31) == 0) atomicAdd(acc, per);
}

// ---------------------------------------------------------------- kNN + loss

// sorted ascending top-K insert; all indices static after unroll -> registers
__device__ __forceinline__ void topk_insert(float hd[KNN], int hi[KNN], float d, int j) {
    if (d >= hd[KNN - 1]) return;
    float cd = d; int ci = j;
#pragma unroll
    for (int t = 0; t < KNN; ++t) {
        bool sw = cd < hd[t];
        float td = hd[t]; int ti = hi[t];
        hd[t] = sw ? cd : td;
        hi[t] = sw ? ci : ti;
        cd = sw ? td : cd;
        ci = sw ? ti : ci;
    }
}

// One wave (32 lanes) per 16-row tile.
// WMMA f32 16x16x4 layouts (ISA 7.12.2):
//   A 16x4 : lane L, vgpr v -> A[M = L%16][k = 2*(L/16) + v]
//   B 4x16 : lane L, vgpr v -> B[k = 2*(L/16) + v][N = L%16]
//   C 16x16: lane L, vgpr r -> C[M = r + 8*(L/16)][N = L%16]
// Gram = E * E^T, so A and B tiles load identically: float2 at
// emb[(base + L%16)*16 + 4c + 2*(L/16)].
__global__ __launch_bounds__(32) void knn_kernel(const float* __restrict__ emb,
                                                 const int* __restrict__ pid,
                                                 const float* __restrict__ sq,
                                                 float* __restrict__ acc) {
    __shared__ float tile[16 * 17];

    const int lane = threadIdx.x;     // 0..31
    const int n    = lane & 15;
    const int h    = lane >> 4;       // half-wave: 0 or 1
    const int i0   = blockIdx.x * 16;
    const int rowg = i0 + n;          // row this lane scans

    // A tile (rows i0..i0+15), resident for the whole pass
    v2f a[4];
#pragma unroll
    for (int c = 0; c < 4; ++c)
        a[c] = *(const v2f*)(emb + (i0 + n) * DIM + 4 * c + 2 * h);

    // squared norms of the 8 rows this lane's accumulator covers (M = r + 8h)
    float sqI[8];
#pragma unroll
    for (int r = 0; r < 8; ++r)
        sqI[r] = sq[i0 + r + 8 * h];

    float hd[KNN]; int hi[KNN];
#pragma unroll
    for (int t = 0; t < KNN; ++t) { hd[t] = 3.4e38f; hi[t] = 0; }

    for (int j0 = 0; j0 < N_NODES; j0 += 16) {
        // B tile (cols j0..j0+15): same layout as A (Gram symmetry)
        v2f b[4];
#pragma unroll
        for (int c = 0; c < 4; ++c)
            b[c] = *(const v2f*)(emb + (j0 + n) * DIM + 4 * c + 2 * h);

        // 16x16 fp32 Gram tile: 4 chained K=4 WMMAs
        v8f g = {};
#pragma unroll
        for (int c = 0; c < 4; ++c)
            g = __builtin_amdgcn_wmma_f32_16x16x4_f32(
                    false, a[c], false, b[c], (short)0, g, false, false);

        // d2 = sq_i + sq_j - 2*gram, spill tile to LDS transposed-access-friendly
        float sqJ = sq[j0 + n];
#pragma unroll
        for (int r = 0; r < 8; ++r) {
            int M = r + 8 * h;
            tile[M * 17 + n] = sqI[r] + sqJ - 2.0f * g[r];
        }
        __syncthreads();

        // scan: lane owns row n; half-wave h scans columns h*8 .. h*8+7
#pragma unroll
        for (int k = 0; k < 8; ++k) {
            int col = h * 8 + k;
            int j   = j0 + col;
            float d2 = tile[n * 17 + col];
            if (j != rowg) topk_insert(hd, hi, d2, j);
        }
        __syncthreads();
    }

    // merge the partner half-wave's top-K (snapshot first, then insert)
    float od[KNN]; int oi[KNN];
#pragma unroll
    for (int t = 0; t < KNN; ++t) {
        od[t] = __shfl_xor(hd[t], 16, 32);
        oi[t] = __shfl_xor(hi[t], 16, 32);
    }
#pragma unroll
    for (int t = 0; t < KNN; ++t) topk_insert(hd, hi, od[t], oi[t]);

    // hinge contributions for this row's K neighbors (half-wave 0 only)
    float sum = 0.f;
    if (h == 0) {
        int pi = pid[rowg];
#pragma unroll
        for (int t = 0; t < KNN; ++t) {
            int j   = hi[t];
            float d = sqrtf(fmaxf(hd[t], 0.f) + EPSF);
            bool y  = (pid[j] == pi) && (pi != 0);
            sum += y ? d : fmaxf(0.f, MARGIN - d);
        }
    }
#pragma unroll
    for (int off = 16; off > 0; off >>= 1)
        sum += __shfl_down(sum, off, 32);
    if (lane == 0) atomicAdd(acc, sum);
}

// ---------------------------------------------------------------- finalize

__global__ void finalize_kernel(const float* __restrict__ acc, float* __restrict__ out) {
    float sig = acc[0] / (float)N_SIG;
    float knn = acc[1] / (float)(N_NODES * KNN);
    float rnd = acc[2] / (float)N_RAND;
    out[0] = sig + knn + rnd;
    out[1] = sig;
    out[2] = knn;
    out[3] = rnd;
}

// ---------------------------------------------------------------- launch

extern "C" void kernel_launch(void* const* d_in, const int* in_sizes, int n_in,
                              void* d_out, int out_size, void* d_ws, size_t ws_size,
                              hipStream_t stream) {
    const float* emb = (const float*)d_in[0];            // [8192,16] f32
    const int*   pid = (const int*)d_in[1];              // [8192] i32
    const int*   te  = (const int*)d_in[2];              // [2,40000] i32
    const int*   re  = (const int*)d_in[3];              // [2,100000] i32
    float*       out = (float*)d_out;                    // 4 f32

    // ws layout: [0..3] accumulators (sig, knn, rand, pad), [16..] sq[8192]
    float* accw = (float*)d_ws;
    float* sq   = accw + 16;

    init_acc_kernel<<<1, 32, 0, stream>>>(accw);
    sq_kernel<<<N_NODES / 256, 256, 0, stream>>>(emb, sq);

    edge_loss_kernel<<<(N_SIG + 255) / 256, 256, 0, stream>>>(
        emb, pid, te, te + N_SIG, N_SIG, /*force_pos=*/1, accw + 0);
    edge_loss_kernel<<<(N_RAND + 255) / 256, 256, 0, stream>>>(
        emb, pid, re, re + N_RAND, N_RAND, /*force_pos=*/0, accw + 2);

    knn_kernel<<<N_NODES / 16, 32, 0, stream>>>(emb, pid, sq, accw + 1);

    finalize_kernel<<<1, 1, 0, stream>>>(accw, out);
}